// RGBBlock_57664230916824
// MI455X (gfx1250) — compile-verified
//
#include <hip/hip_runtime.h>

typedef __attribute__((ext_vector_type(2))) float v2f;
typedef __attribute__((ext_vector_type(4))) float v4f;
typedef __attribute__((ext_vector_type(8))) float v8f;

#define B_DIM   8
#define C_IN    256
#define L_DIM   512
#define H_DIM   128
#define W_DIM   128
#define HW      (H_DIM * W_DIM)
#define HALF_H  64
#define HALF_W  64
#define OC      3
#define M_PAD   16

// ---------------------------------------------------------------------------
// Kernel 1: style = istyle @ style_w^T + style_b ; build zero-padded modulated
// weight matrices Wpad[b][16][256] = conv_w[o][c] * (style[b][c] + 1), rows
// o>=3 zeroed.  8 blocks (one per batch) x 256 threads (one per channel c).
// ---------------------------------------------------------------------------
__global__ void build_weights(const float* __restrict__ istyle,
                              const float* __restrict__ style_w,
                              const float* __restrict__ style_b,
                              const float* __restrict__ conv_w,
                              float* __restrict__ wpad) {
    const int b = blockIdx.x;
    const int c = threadIdx.x;
    const float* irow = istyle + b * L_DIM;
    const float* wrow = style_w + c * L_DIM;
    float acc = 0.0f;
    #pragma unroll 4
    for (int l = 0; l < L_DIM; l += 4) {
        v4f a = *(const v4f*)(irow + l);
        v4f w = *(const v4f*)(wrow + l);
        acc += a.x * w.x + a.y * w.y + a.z * w.z + a.w * w.w;
    }
    const float s1 = acc + style_b[c] + 1.0f;   // style + 1
    float* dst = wpad + b * (M_PAD * C_IN);
    #pragma unroll
    for (int o = 0; o < M_PAD; ++o) {
        float v = (o < OC) ? conv_w[o * C_IN + c] * s1 : 0.0f;
        dst[o * C_IN + c] = v;                  // coalesced across c
    }
}

// ---------------------------------------------------------------------------
// Kernel 2: fused modulated 1x1 conv (WMMA f32 16x16x4, M padded 3->16) +
// bias + 2x bilinear (half-pixel) upsample-add of prev_rgb.
// Grid: 1024 blocks = 8 batches * 128 output rows. Block: 256 thr = 8 waves,
// each wave owns 16 contiguous pixels of the row, K = 256 reduction.
// ---------------------------------------------------------------------------
__global__ void __launch_bounds__(256)
rgb_mod_conv(const float* __restrict__ x,
             const float* __restrict__ prev_rgb,
             const float* __restrict__ wpad,
             const float* __restrict__ rgb_bias,
             float* __restrict__ out) {
    __shared__ float lds_w[M_PAD * C_IN];       // 16 KB: padded A-matrix

    const int b    = blockIdx.x >> 7;           // batch
    const int row  = blockIdx.x & 127;          // output row h
    const int tid  = threadIdx.x;

    // Cooperative stage of Wpad[b] (1024 float4) into LDS.
    {
        const v4f* s4 = (const v4f*)(wpad + b * (M_PAD * C_IN));
        v4f* d4 = (v4f*)lds_w;
        #pragma unroll
        for (int i = 0; i < 4; ++i)
            d4[tid + 256 * i] = s4[tid + 256 * i];
    }
    __syncthreads();

    const int wave = tid >> 5;
    const int lane = tid & 31;
    const int half = lane >> 4;                 // lane half selects K pair
    const int ln   = lane & 15;
    const int wcol = wave * 16 + ln;            // pixel column 0..127
    const int p    = row * W_DIM + wcol;        // flat pixel in HxW

    // Per-lane x base: lanes 0-15 feed B rows {k, k+1}; lanes 16-31 {k+2, k+3}
    const float* xlane = x + (size_t)b * C_IN * HW + (size_t)(half * 2) * HW + p;

    v8f acc = {};
    #pragma unroll 4
    for (int k0 = 0; k0 < C_IN; k0 += 4) {
        // A fragment (16x4 f32): lane ln row, K = k0 + 2*half + {0,1}
        const int kA = k0 + half * 2;
        v2f afrag = *(const v2f*)(lds_w + ln * C_IN + kA);   // ds_load_b64
        // B fragment (4x16 f32): VGPR0 = rows {k0, k0+2}, VGPR1 = {k0+1, k0+3}
        v2f bfrag;
        bfrag.x = xlane[k0 * HW];                            // global_load_b32
        bfrag.y = xlane[k0 * HW + HW];                       // global_load_b32
        acc = __builtin_amdgcn_wmma_f32_16x16x4_f32(
            /*neg_a=*/false, afrag, /*neg_b=*/false, bfrag,
            /*c_mod=*/(short)0, acc, /*reuse_a=*/false, /*reuse_b=*/false);
    }

    // D layout: VGPR r, lanes 0-15 -> (M=r, N=ln). Channels 0..2 live in
    // acc[0..2] of lanes 0-15; padded rows are discarded.
    if (half == 0) {
        // Half-pixel bilinear source coords for 2x upsample.
        const float fy = row  * 0.5f - 0.25f;
        const float fx = wcol * 0.5f - 0.25f;
        const float y0f = floorf(fy), x0f = floorf(fx);
        const float wy = fy - y0f,   wx = fx - x0f;
        const int iy0 = min(max((int)y0f,     0), HALF_H - 1);
        const int iy1 = min(max((int)y0f + 1, 0), HALF_H - 1);
        const int ix0 = min(max((int)x0f,     0), HALF_W - 1);
        const int ix1 = min(max((int)x0f + 1, 0), HALF_W - 1);
        const float* pb = prev_rgb + (size_t)b * OC * HALF_H * HALF_W;
        #pragma unroll
        for (int o = 0; o < OC; ++o) {
            const float* po = pb + o * HALF_H * HALF_W;
            const float v00 = po[iy0 * HALF_W + ix0];
            const float v01 = po[iy0 * HALF_W + ix1];
            const float v10 = po[iy1 * HALF_W + ix0];
            const float v11 = po[iy1 * HALF_W + ix1];
            const float up = (1.0f - wy) * ((1.0f - wx) * v00 + wx * v01)
                           +         wy  * ((1.0f - wx) * v10 + wx * v11);
            out[((size_t)b * OC + o) * HW + p] = acc[o] + rgb_bias[o] + up;
        }
    }
}

// ---------------------------------------------------------------------------
extern "C" void kernel_launch(void* const* d_in, const int* in_sizes, int n_in,
                              void* d_out, int out_size, void* d_ws, size_t ws_size,
                              hipStream_t stream) {
    const float* x        = (const float*)d_in[0];
    const float* prev_rgb = (const float*)d_in[1];
    const float* istyle   = (const float*)d_in[2];
    const float* style_w  = (const float*)d_in[3];
    const float* style_b  = (const float*)d_in[4];
    const float* conv_w   = (const float*)d_in[5];
    const float* rgb_bias = (const float*)d_in[6];
    float* out  = (float*)d_out;
    float* wpad = (float*)d_ws;                 // 8*16*256*4 = 128 KB scratch

    build_weights<<<B_DIM, C_IN, 0, stream>>>(istyle, style_w, style_b, conv_w, wpad);
    rgb_mod_conv<<<B_DIM * H_DIM, 256, 0, stream>>>(x, prev_rgb, wpad, rgb_bias, out);
}